// CenterLoss_45681272160610
// MI455X (gfx1250) — compile-verified
//
#include <hip/hip_runtime.h>
#include <hip/hip_bf16.h>

typedef __attribute__((ext_vector_type(2))) float v2f;
typedef __attribute__((ext_vector_type(8))) float v8f;

#define FEAT_DIM    512
#define BATCH       8192
#define ROWS_TILE   16      // rows per WMMA tile (one wave)
#define WAVES_BLK   8       // waves per block, each takes a K-slice
#define K_SLICE     (FEAT_DIM / WAVES_BLK)   // 64 dims per wave

__global__ void CenterLoss_zero_kernel(float* out) { out[0] = 0.0f; }

// Each block: 16 feature rows. Each of the 8 waves handles a 64-dim K-slice.
// Per wave: accumulate Gram(d) with V_WMMA_F32_16X16X4_F32 where d = f - c_label,
// then pull the diagonal (per-row squared norms) and reduce.
__global__ __launch_bounds__(256)
void CenterLoss_45681272160610_kernel(const float* __restrict__ features,
                                      const int*   __restrict__ labels,
                                      const float* __restrict__ centers,
                                      float*       __restrict__ out) {
    const int lane      = threadIdx.x & 31;
    const int wave      = threadIdx.x >> 5;
    const int rowInTile = lane & 15;
    const int hi        = lane >> 4;            // 0 -> K={k,k+1}, 1 -> K={k+2,k+3}
    const int row       = blockIdx.x * ROWS_TILE + rowInTile;

    const int label = labels[row];

    // Base pointers: A-matrix 16x4 f32 layout wants lane l<16 to hold K=k,k+1
    // and lane l+16 to hold K=k+2,k+3 of the same row -> offset by hi*2.
    const float* __restrict__ f =
        features + (size_t)row * FEAT_DIM + (size_t)wave * K_SLICE + hi * 2;
    const float* __restrict__ c =
        centers + (size_t)label * FEAT_DIM + (size_t)wave * K_SLICE + hi * 2;

    v8f acc = {};
    #pragma unroll
    for (int k = 0; k < K_SLICE; k += 4) {
        v2f fv = *(const v2f*)(f + k);          // global_load_b64
        v2f cv = *(const v2f*)(c + k);          // gathered center row
        v2f d  = fv - cv;
        // D = d * d^T + C : diagonal accumulates per-row squared norms (f32)
        acc = __builtin_amdgcn_wmma_f32_16x16x4_f32(
                  /*neg_a=*/false, d, /*neg_b=*/false, d,
                  /*c_mod=*/(short)0, acc, /*reuse_a=*/false, /*reuse_b=*/false);
    }

    // Extract Gram diagonal from the 16x16 f32 C/D layout:
    //   (m,m) m<8  -> lane m,     VGPR m
    //   (m,m) m>=8 -> lane m+16,  VGPR m-8  (lanes 24..31, VGPR lane-24)
    float part = 0.0f;
    if (lane < 8)        part = acc[lane];
    else if (lane >= 24) part = acc[lane - 24];

    // wave32 reduction
    #pragma unroll
    for (int off = 16; off > 0; off >>= 1)
        part += __shfl_xor(part, off, 32);

    __shared__ float spart[WAVES_BLK];
    if (lane == 0) spart[wave] = part;
    __syncthreads();

    if (threadIdx.x == 0) {
        float s = 0.0f;
        #pragma unroll
        for (int i = 0; i < WAVES_BLK; ++i) s += spart[i];
        atomicAdd(out, s * (1.0f / (float)BATCH));
    }
}

extern "C" void kernel_launch(void* const* d_in, const int* in_sizes, int n_in,
                              void* d_out, int out_size, void* d_ws, size_t ws_size,
                              hipStream_t stream) {
    const float* features = (const float*)d_in[0];
    const int*   labels   = (const int*)d_in[1];
    const float* centers  = (const float*)d_in[2];
    float*       out      = (float*)d_out;

    CenterLoss_zero_kernel<<<1, 1, 0, stream>>>(out);

    const int blocks = BATCH / ROWS_TILE;   // 512 blocks x 256 threads = 4096 waves
    CenterLoss_45681272160610_kernel<<<blocks, 256, 0, stream>>>(
        features, labels, centers, out);
}